// TFF_RepulsionCG_76020921139248
// MI455X (gfx1250) — compile-verified
//
#include <hip/hip_runtime.h>

#define N_ATOMS 4096
#define N3 (N_ATOMS * 3)          // 12288 force components
#define CUTOFF 12.0f

typedef __attribute__((ext_vector_type(2))) float v2f;
typedef __attribute__((ext_vector_type(8))) float v8f;

// d_out layout: [0] = energy, [1 .. 12288] = forces (4096 x 3 flattened)

__global__ void TFF_Repulsion_init(const float* __restrict__ forces_in,
                                   float* __restrict__ out) {
    int i = blockIdx.x * blockDim.x + threadIdx.x;
    if (i == 0) out[0] = 0.0f;
    if (i < N3) out[1 + i] = forces_in[i];
}

__global__ __launch_bounds__(256) void TFF_Repulsion_pairs(
    const float* __restrict__ dist,   // [4096*4096]
    const float* __restrict__ vec,    // [4096*4096*3]
    const float* __restrict__ rep,    // [n_pairs]
    const int*   __restrict__ coord,  // [n_pairs*2]
    int n_pairs,
    float* __restrict__ out) {        // [1 + 12288]
    // Full force image lives in LDS (48 KB of the 320 KB WGP pool):
    // 24M global atomics become LDS ds_add_f32 + one flush per block.
    __shared__ float s_forces[N3];
    for (int j = threadIdx.x; j < N3; j += blockDim.x) s_forces[j] = 0.0f;
    __syncthreads();

    float e0 = 0.0f, e1 = 0.0f;       // two accumulators -> fills both A VGPRs
    int idx    = blockIdx.x * blockDim.x + threadIdx.x;
    int stride = gridDim.x * blockDim.x;
    int t = 0;
    for (int i = idx; i < n_pairs; i += stride) {
        int a = coord[2 * i];
        int b = coord[2 * i + 1];
        int off = (a << 12) + b;      // a*4096 + b
        float d = dist[off];
        if (d <= CUTOFF) {
            float B    = rep[i];
            float inv  = 1.0f / d;
            float inv2 = inv * inv;
            float inv6 = inv2 * inv2 * inv2;
            float e    = B * inv6;
            if (t) e1 += e; else e0 += e;
            float f = -6.0f * e * inv;            // -6*B*inv^7
            const float* v = vec + (size_t)off * 3u;
            float fx = f * v[0], fy = f * v[1], fz = f * v[2];
            atomicAdd(&s_forces[3 * a + 0],  fx);
            atomicAdd(&s_forces[3 * a + 1],  fy);
            atomicAdd(&s_forces[3 * a + 2],  fz);
            atomicAdd(&s_forces[3 * b + 0], -fx);
            atomicAdd(&s_forces[3 * b + 1], -fy);
            atomicAdd(&s_forces[3 * b + 2], -fz);
        }
        t ^= 1;
    }
    __syncthreads();

    // Flush block-local force image to global (distributed atomics).
    for (int j = threadIdx.x; j < N3; j += blockDim.x)
        atomicAdd(&out[1 + j], s_forces[j]);

    // Wave-level energy reduction via f32 WMMA (exact, RNE):
    // A = 16x4 holding all 64 lane-accumulators, B = 4x16 ones.
    // D[m][n] = sum_k A[m][k]; summing one column over both lane halves
    // yields the grand total regardless of A's lane->(m,k) mapping.
    v2f a2; a2[0] = e0;   a2[1] = e1;
    v2f b2; b2[0] = 1.0f; b2[1] = 1.0f;
    v8f c = {};
    c = __builtin_amdgcn_wmma_f32_16x16x4_f32(
            /*neg_a=*/false, a2, /*neg_b=*/false, b2,
            /*c_mod=*/(short)0, c, /*reuse_a=*/false, /*reuse_b=*/false);
    float s = c[0] + c[1] + c[2] + c[3] + c[4] + c[5] + c[6] + c[7];
    s += __shfl_xor(s, 16, 32);       // lane n + lane n+16 = grand total (wave32)
    if ((threadIdx.x & 31) == 0)
        atomicAdd(&out[0], s);        // one global atomic per wave
}

extern "C" void kernel_launch(void* const* d_in, const int* in_sizes, int n_in,
                              void* d_out, int out_size, void* d_ws, size_t ws_size,
                              hipStream_t stream) {
    const float* dist      = (const float*)d_in[0];
    const float* vec       = (const float*)d_in[1];
    const float* forces_in = (const float*)d_in[2];
    const float* rep       = (const float*)d_in[3];
    const int*   coord     = (const int*)d_in[4];
    float* out = (float*)d_out;
    int n_pairs = in_sizes[3];

    int initThreads = 256;
    int initBlocks  = (N3 + 1 + initThreads - 1) / initThreads;
    TFF_Repulsion_init<<<initBlocks, initThreads, 0, stream>>>(forces_in, out);

    // 512 blocks x 256 threads: ~30 pairs/thread, 48 KB LDS each
    // (<= 6 resident blocks/WGP), flush traffic 512*12288 ~ 6M atomics.
    TFF_Repulsion_pairs<<<512, 256, 0, stream>>>(dist, vec, rep, coord, n_pairs, out);
}